// ComplEx_8564164788315
// MI455X (gfx1250) — compile-verified
//
#include <hip/hip_runtime.h>

typedef __attribute__((ext_vector_type(2))) float v2f;
typedef __attribute__((ext_vector_type(8))) float v8f;

#define HIDDEN 128
#define ZDIM 256

// One wave handles 16 edges. score(e) = c_e . r_e with
//   c_e[k]       = hr*tr + hi*ti   (k < 128)
//   c_e[128+k]   = hr*ti - hi*tr
//   r_e[k]       = rel_real[type][k],  r_e[128+k] = rel_imag[type][k]
// Computed as diag( A(16x256) x B(256x16) ) via 64 V_WMMA_F32_16X16X4_F32.
__global__ __launch_bounds__(256) void ComplEx_score_wmma(
    const float* __restrict__ z,
    const int*   __restrict__ edge_src,
    const int*   __restrict__ edge_dst,
    const int*   __restrict__ edge_type,
    const float* __restrict__ rel_re,
    const float* __restrict__ rel_im,
    float*       __restrict__ out,
    int num_edges)
{
    const int lane = threadIdx.x & 31;
    const int wave = threadIdx.x >> 5;
    const int e0   = (blockIdx.x * (blockDim.x >> 5) + wave) * 16;
    if (e0 >= num_edges) return;          // wave-uniform: EXEC stays all-1s

    const int m    = lane & 15;           // edge row (A) / edge col (B) for this lane
    const int half = lane >> 4;           // K sub-offset: lanes 0-15 -> K{0,1}, 16-31 -> K{2,3}
    int em = e0 + m;
    if (em >= num_edges) em = num_edges - 1;   // clamp (keeps all lanes active)

    const int src = edge_src[em];
    const int dst = edge_dst[em];
    const int t   = edge_type[em];

    const float* hp  = z      + (size_t)src * ZDIM;
    const float* tp  = z      + (size_t)dst * ZDIM;
    const float* rrp = rel_re + (size_t)t   * HIDDEN;
    const float* rip = rel_im + (size_t)t   * HIDDEN;

    const int koff = 2 * half;            // lane's K offset inside each 4-wide chunk

    v8f acc = {};                         // 16x16 f32 accumulator (8 VGPRs)

#pragma unroll 4
    for (int it = 0; it < HIDDEN / 4; ++it) {
        const int k = 4 * it + koff;      // dim pair {k, k+1} for this lane

        v2f hr = *(const v2f*)(hp + k);
        v2f hi = *(const v2f*)(hp + HIDDEN + k);
        v2f tr = *(const v2f*)(tp + k);
        v2f ti = *(const v2f*)(tp + HIDDEN + k);
        v2f rr = *(const v2f*)(rrp + k);
        v2f ri = *(const v2f*)(rip + k);

        v2f a, b;
        a.x = hr.x * tr.x + hi.x * ti.x;
        a.y = hr.y * tr.y + hi.y * ti.y;
        b.x = hr.x * ti.x - hi.x * tr.x;
        b.y = hr.y * ti.y - hi.y * tr.y;

        // D = A*B + C ; full-f32 path (no precision loss vs reference)
        acc = __builtin_amdgcn_wmma_f32_16x16x4_f32(
            false, a, false, rr, (short)0, acc, false, false);
        acc = __builtin_amdgcn_wmma_f32_16x16x4_f32(
            false, b, false, ri, (short)0, acc, false, false);
    }

    // Diagonal extraction per the 16x16 f32 C/D layout:
    //   edge m<8  : VGPR m,   lane m        (M=m, N=m)
    //   edge m>=8 : VGPR m-8, lane m+16     (M=v+8, N=lane-16)
    int om = -1, vi = 0;
    if (lane < 8)        { om = lane;      vi = lane;      }
    else if (lane >= 24) { om = lane - 16; vi = lane - 24; }

    if (om >= 0) {
        const int eo = e0 + om;
        if (eo < num_edges) {
            float s;
            switch (vi) {
                case 0: s = acc[0]; break;
                case 1: s = acc[1]; break;
                case 2: s = acc[2]; break;
                case 3: s = acc[3]; break;
                case 4: s = acc[4]; break;
                case 5: s = acc[5]; break;
                case 6: s = acc[6]; break;
                default: s = acc[7]; break;
            }
            out[eo] = s;
        }
    }
}

extern "C" void kernel_launch(void* const* d_in, const int* in_sizes, int n_in,
                              void* d_out, int out_size, void* d_ws, size_t ws_size,
                              hipStream_t stream) {
    const float* z          = (const float*)d_in[0];
    const int*   edge_index = (const int*)  d_in[1];   // [2, E] flat: src then dst
    const int*   edge_type  = (const int*)  d_in[2];
    const float* rel_re     = (const float*)d_in[3];
    const float* rel_im     = (const float*)d_in[4];
    float*       out        = (float*)d_out;

    const int E = in_sizes[2];             // NUM_EDGES (edge_type count)
    const int* src = edge_index;
    const int* dst = edge_index + E;

    const int waves_needed    = (E + 15) / 16;
    const int waves_per_block = 8;         // 256 threads, wave32
    const int blocks = (waves_needed + waves_per_block - 1) / waves_per_block;

    ComplEx_score_wmma<<<blocks, 256, 0, stream>>>(
        z, src, dst, edge_type, rel_re, rel_im, out, E);
}